// SSIMLoss3D_42846593744948
// MI455X (gfx1250) — compile-verified
//
#include <hip/hip_runtime.h>
#include <cmath>

// Fused separable 3D-SSIM for MI455X (gfx1250, wave32).
// x-axis 7-tap conv = Toeplitz matmul on V_WMMA_F32_16X16X4_F32 (exact f32);
// y/z convs are VALU FMAs (LDS + register ring). Memory-bound: ~230MB traffic.

typedef __attribute__((ext_vector_type(2))) float v2f;
typedef __attribute__((ext_vector_type(8))) float v8f;

#define DIMS    192
#define ODIM    186            // 192 - 7 + 1
#define TS      16             // output tile (y,x)
#define HALO    22             // TS + 6
#define PSTRIDE 24             // plane row stride; K padded 22 -> 24 for 6 WMMA K-steps
#define ZCHUNK  64
#define NPLANES (ZCHUNK + 6)

struct W7 { float w[7]; };

// One (field, row-group) x-conv unit: S1[16x16] += Field[16x4] * Wband[4x16], 6 K-steps.
// Branch-free field synthesis: A = (pa*a + pb*b) * (q1 + qa*a + qb*b)
//   f=0: a   f=1: b   f=2: a*a   f=3: b*b   f=4: a*b
__device__ __forceinline__ void xconv_unit(
    const int u, const int lane,
    const float* __restrict__ shA, const float* __restrict__ shB,
    float* __restrict__ s1,            // [5][HALO*TS] flattened
    const v2f* __restrict__ Bf)
{
    const int f = u >> 1;
    const int g = u & 1;

    const float pa = (f == 1 || f == 3) ? 0.0f : 1.0f;
    const float pb = 1.0f - pa;
    const float q1 = (f < 2)  ? 1.0f : 0.0f;
    const float qa = (f == 2) ? 1.0f : 0.0f;
    const float qb = (f >= 3) ? 1.0f : 0.0f;

    int row = (lane & 15) + (g << 4);
    if (row > HALO - 1) row = HALO - 1;          // junk rows map to unused C rows
    const int kb = (lane >> 4) << 1;             // A layout: lanes 16-31 hold K+2

    v8f accu = {0.f, 0.f, 0.f, 0.f, 0.f, 0.f, 0.f, 0.f};
#pragma unroll
    for (int s = 0; s < 6; ++s) {
        const int k0 = s * 4 + kb;
        const float a0 = shA[row * PSTRIDE + k0];
        const float a1 = shA[row * PSTRIDE + k0 + 1];
        const float b0 = shB[row * PSTRIDE + k0];
        const float b1 = shB[row * PSTRIDE + k0 + 1];
        v2f A;
        A[0] = fmaf(pa, a0, pb * b0) * fmaf(qa, a0, fmaf(qb, b0, q1));
        A[1] = fmaf(pa, a1, pb * b1) * fmaf(qa, a1, fmaf(qb, b1, q1));
        accu = __builtin_amdgcn_wmma_f32_16x16x4_f32(
                   false, A, false, Bf[s], (short)0, accu, false, false);
    }
    // C layout: lane L, vgpr v -> row v + 8*(L>=16), col L&15
    const int n = lane & 15;
#pragma unroll
    for (int v = 0; v < 8; ++v) {
        const int m = v + ((lane >> 4) << 3) + (g << 4);
        if (m < HALO) s1[f * (HALO * TS) + m * TS + n] = accu[v];
    }
}

__global__ __launch_bounds__(256)
void ssim3d_fused(const float* __restrict__ inA,
                  const float* __restrict__ inB,
                  double* __restrict__ acc,
                  W7 wp)
{
    __shared__ float shA[HALO * PSTRIDE];   // input plane tile  (22 x 24)
    __shared__ float shB[HALO * PSTRIDE];   // target plane tile
    __shared__ float s1[5 * HALO * TS];     // after x-conv: 5 fields x 22 rows x 16 cols
    __shared__ float red[256];

    const int tid  = threadIdx.x;
    const int lane = tid & 31;
    const int wv   = tid >> 5;              // 8 waves / block

    const int tileIdx = blockIdx.x;         // 0..143 (12 x 12 tiles cover 186)
    const int x0 = (tileIdx % 12) * TS;
    const int y0 = (tileIdx / 12) * TS;
    const int z0 = blockIdx.y * ZCHUNK;
    const int batch = blockIdx.z;
    const size_t batchOff = (size_t)batch * DIMS * DIMS * DIMS;

    float w[7];
#pragma unroll
    for (int t = 0; t < 7; ++t) w[t] = wp.w[t];

    // B fragments: banded Toeplitz W[k,n] = w[k-n] (0<=k-n<7), K=24 in 6 slices.
    // Layout (f32 4x16, mirroring A 16x4): lanes 0-15 hold K rows {0,1}, lanes 16-31 rows {2,3}.
    v2f Bf[6];
    {
        const int n = lane & 15;
#pragma unroll
        for (int s = 0; s < 6; ++s) {
#pragma unroll
            for (int j = 0; j < 2; ++j) {
                const int k = s * 4 + ((lane >> 4) << 1) + j;
                const int t = k - n;
                Bf[s][j] = (t >= 0 && t < 7) ? w[t] : 0.0f;
            }
        }
    }

    // z ring buffer (7 planes of post-xy-conv values) in registers, 1 voxel / thread
    float ring[5][7];
#pragma unroll
    for (int f = 0; f < 5; ++f)
#pragma unroll
        for (int t = 0; t < 7; ++t) ring[f][t] = 0.0f;

    const int ty = tid >> 4;                // voxel within tile
    const int tx = tid & 15;

    const float c1 = 1.0e-4f;               // (K1*MAX_VAL)^2
    const float c2 = 9.0e-4f;               // (K2*MAX_VAL)^2
    float lsum = 0.0f;

    for (int p = 0; p < NPLANES; ++p) {
        const int z = z0 + p;

        // ---- cooperative plane load (guarded, OOB -> 0; cols 22/23 hit zero B weights) ----
        for (int i = tid; i < HALO * PSTRIDE; i += 256) {
            const int r  = i / PSTRIDE;
            const int c  = i % PSTRIDE;
            const int gy = y0 + r;
            const int gx = x0 + c;
            float a = 0.0f, b = 0.0f;
            if (z < DIMS && gy < DIMS && gx < DIMS) {
                const size_t idx = batchOff + ((size_t)z * DIMS + gy) * DIMS + gx;
                a = inA[idx];
                b = inB[idx];
            }
            shA[i] = a;
            shB[i] = b;
        }

        // ---- speculative prefetch of next plane (global_prefetch_b8) ----
        {
            const int zn = z + 1;
            if (zn < DIMS && tid < 2 * HALO) {
                const int r   = tid >> 1;
                const int gx0 = x0 + (tid & 1) * 16;     // 2 cachelines cover 24-col row
                const int gy  = y0 + r;
                if (gy < DIMS && gx0 < DIMS) {
                    const size_t idx = batchOff + ((size_t)zn * DIMS + gy) * DIMS + gx0;
                    __builtin_prefetch(inA + idx, 0, 1);
                    __builtin_prefetch(inB + idx, 0, 1);
                }
            }
        }
        __syncthreads();

        // ---- x-conv via WMMA: 10 units = 5 fields x 2 row-groups over 8 waves ----
        xconv_unit(wv, lane, shA, shB, s1, Bf);
        if (wv < 2) xconv_unit(wv + 8, lane, shA, shB, s1, Bf);
        __syncthreads();

        // ---- y-conv per voxel, push into z ring ----
#pragma unroll
        for (int f = 0; f < 5; ++f) {
            float a = 0.0f;
#pragma unroll
            for (int t = 0; t < 7; ++t)
                a = fmaf(w[t], s1[f * (HALO * TS) + (ty + t) * TS + tx], a);
#pragma unroll
            for (int t = 0; t < 6; ++t) ring[f][t] = ring[f][t + 1];
            ring[f][6] = a;
        }

        // ---- z-conv + SSIM once the ring holds 7 planes ----
        if (p >= 6) {
            const int oz = z - 6;
            if (oz < ODIM && (y0 + ty) < ODIM && (x0 + tx) < ODIM) {
                float m1 = 0.f, m2 = 0.f, exx = 0.f, eyy = 0.f, exy = 0.f;
#pragma unroll
                for (int t = 0; t < 7; ++t) {
                    m1  = fmaf(w[t], ring[0][t], m1);
                    m2  = fmaf(w[t], ring[1][t], m2);
                    exx = fmaf(w[t], ring[2][t], exx);
                    eyy = fmaf(w[t], ring[3][t], eyy);
                    exy = fmaf(w[t], ring[4][t], exy);
                }
                const float mu11 = m1 * m1, mu22 = m2 * m2, mu12 = m1 * m2;
                const float s11 = exx - mu11;
                const float s22 = eyy - mu22;
                const float s12 = exy - mu12;
                const float num = (2.f * mu12 + c1) * (2.f * s12 + c2);
                const float den = (mu11 + mu22 + c1) * (s11 + s22 + c2);
                lsum += num / den;
            }
        }
        __syncthreads();    // protect shA/shB/s1 before next iteration overwrites
    }

    // ---- block reduction + global accumulate (f64: 12.9M terms) ----
    red[tid] = lsum;
    __syncthreads();
    for (int s = 128; s > 0; s >>= 1) {
        if (tid < s) red[tid] += red[tid + s];
        __syncthreads();
    }
    if (tid == 0) atomicAdd(acc, (double)red[0]);
}

__global__ void ssim3d_final(const double* __restrict__ acc, float* __restrict__ out)
{
    out[0] = (float)(acc[0] / (double)(2ull * 186ull * 186ull * 186ull));
}

extern "C" void kernel_launch(void* const* d_in, const int* in_sizes, int n_in,
                              void* d_out, int out_size, void* d_ws, size_t ws_size,
                              hipStream_t stream)
{
    const float* inA = (const float*)d_in[0];   // 'input'
    const float* inB = (const float*)d_in[1];   // 'target'
    double* acc = (double*)d_ws;
    float*  out = (float*)d_out;

    // 1D normalized Gaussian taps: softmax over 7^3 of (g_i+g_j+g_k) factorizes
    // into the outer product of these 1D softmax weights.
    W7 wp;
    {
        double g[7], s = 0.0;
        for (int i = 0; i < 7; ++i) {
            const double d = (double)i - 3.0;
            g[i] = std::exp(-(d * d) / (2.0 * 1.5 * 1.5));
            s += g[i];
        }
        for (int i = 0; i < 7; ++i) wp.w[i] = (float)(g[i] / s);
    }

    hipMemsetAsync(acc, 0, sizeof(double), stream);     // graph-capture safe

    dim3 grid(12 * 12, (ODIM + ZCHUNK - 1) / ZCHUNK, 2);  // (144, 3, 2)
    ssim3d_fused<<<grid, 256, 0, stream>>>(inA, inB, acc, wp);
    ssim3d_final<<<1, 1, 0, stream>>>(acc, out);
}